// GAT_18571438588582
// MI455X (gfx1250) — compile-verified
//
#include <hip/hip_runtime.h>

// 3-layer GAT forward for MI455X (gfx1250, wave32).
//   - fp32 WMMA (v_wmma_f32_16x16x4_f32) for the 64x64 projections (GEMM is
//     ~2.5 GFLOP total -> negligible; fp32 matrix pipe keeps full fidelity).
//   - Edge phase is the real cost (~410MB gather + ~410MB atomic scatter per
//     layer); working set ~90MB fits MI455X's 192MB L2, so wave-per-edge
//     coalesced 256B row transfers + L2-resident f32 atomics.
//   - Bias b cancels inside BatchNorm ((x+b)-mean(x+b) == x-mean(x)), so it
//     is folded away (it is zero in setup anyway).

#define N_NODES   100000
#define N_EDGES   1600000
#define DIM       64
#define NEG_SLOPE 0.2f
#define BN_EPS    1e-5f

typedef __attribute__((ext_vector_type(2))) float v2f;
typedef __attribute__((ext_vector_type(8))) float v8f;

// ---------------------------------------------------------------------------
// F[nrows,64] = H[nrows,64] @ W[64,64]   (row-major), fp32 WMMA 16x16x4.
// One wave computes a 16x64 tile: 16 k-steps x 4 n-tiles = 64 WMMAs.
// A-frag (16x4 f32): lanes 0-15 hold K=k0..k0+1 (v0,v1), lanes 16-31 K=k0+2..3.
// B-frag (4x16 f32): lane n=lane&15; low half K=k0,k0+1; high half K=k0+2,k0+3.
// C/D (16x16 f32): VGPR r -> rows (r, r+8) split across lane halves.
// ---------------------------------------------------------------------------
__global__ __launch_bounds__(256) void k_gemm64(const float* __restrict__ H,
                                                const float* __restrict__ W,
                                                float* __restrict__ F,
                                                int nrows) {
  __shared__ float Ws[DIM * DIM];
  for (int i = threadIdx.x; i < DIM * DIM; i += 256) Ws[i] = W[i];
  __syncthreads();

  const int wave = threadIdx.x >> 5;
  const int lane = threadIdx.x & 31;
  const int row0 = (blockIdx.x * 8 + wave) * 16;
  if (row0 >= nrows) return;            // wave-uniform: EXEC stays all-1s

  const int m  = lane & 15;
  const int hi = lane >> 4;
  const int kb = hi * 2;
  const float* __restrict__ hrow = H + (size_t)(row0 + m) * DIM;

  v8f acc0 = {}, acc1 = {}, acc2 = {}, acc3 = {};
#pragma unroll
  for (int k0 = 0; k0 < DIM; k0 += 4) {
    v2f a;
    a.x = hrow[k0 + kb];
    a.y = hrow[k0 + kb + 1];
    const float* wr0 = &Ws[(k0 + kb) * DIM];
    const float* wr1 = &Ws[(k0 + kb + 1) * DIM];
    v2f b0; b0.x = wr0[m];      b0.y = wr1[m];
    v2f b1; b1.x = wr0[m + 16]; b1.y = wr1[m + 16];
    v2f b2; b2.x = wr0[m + 32]; b2.y = wr1[m + 32];
    v2f b3; b3.x = wr0[m + 48]; b3.y = wr1[m + 48];
    acc0 = __builtin_amdgcn_wmma_f32_16x16x4_f32(false, a, false, b0, (short)0, acc0, false, false);
    acc1 = __builtin_amdgcn_wmma_f32_16x16x4_f32(false, a, false, b1, (short)0, acc1, false, false);
    acc2 = __builtin_amdgcn_wmma_f32_16x16x4_f32(false, a, false, b2, (short)0, acc2, false, false);
    acc3 = __builtin_amdgcn_wmma_f32_16x16x4_f32(false, a, false, b3, (short)0, acc3, false, false);
  }
#pragma unroll
  for (int r = 0; r < 8; ++r) {
    float* orow = F + (size_t)(row0 + r + hi * 8) * DIM + m;
    orow[0]  = acc0[r];
    orow[16] = acc1[r];
    orow[32] = acc2[r];
    orow[48] = acc3[r];
  }
}

// ---- per-layer re-init: out=0, m=-inf(key), s=0, stats=0 -------------------
__global__ __launch_bounds__(256) void k_init(float* __restrict__ out,
                                              int* __restrict__ mI,
                                              float* __restrict__ s,
                                              float* __restrict__ stats,
                                              int n_nodes) {
  int i = blockIdx.x * 256 + threadIdx.x;
  if (i < n_nodes * DIM) out[i] = 0.0f;
  if (i < n_nodes) { mI[i] = (int)0x80000000; s[i] = 0.0f; }
  if (i < 2 * DIM) stats[i] = 0.0f;
}

// ---- el[i]=feat[i].al, er[i]=feat[i].ar  (one wave32 per row) --------------
__global__ __launch_bounds__(256) void k_elr(const float* __restrict__ F,
                                             const float* __restrict__ al,
                                             const float* __restrict__ ar,
                                             float* __restrict__ el,
                                             float* __restrict__ er,
                                             int n) {
  int row  = blockIdx.x * 8 + (threadIdx.x >> 5);
  int lane = threadIdx.x & 31;
  if (row >= n) return;
  const float* f = F + (size_t)row * DIM;
  float f0 = f[lane], f1 = f[lane + 32];
  float sl = f0 * al[lane] + f1 * al[lane + 32];
  float sr = f0 * ar[lane] + f1 * ar[lane + 32];
#pragma unroll
  for (int off = 16; off > 0; off >>= 1) {
    sl += __shfl_xor(sl, off);
    sr += __shfl_xor(sr, off);
  }
  if (lane == 0) { el[row] = sl; er[row] = sr; }
}

__device__ __forceinline__ int f2ordered(float x) {
  int k = __float_as_int(x);
  return (k < 0) ? (k ^ 0x7fffffff) : k;
}
__device__ __forceinline__ float ordered2f(int k) {
  return __int_as_float((k < 0) ? (k ^ 0x7fffffff) : k);
}

// ---- edge pass 1: logits + per-dst max -------------------------------------
__global__ __launch_bounds__(256) void k_edge_max(const int* __restrict__ src,
                                                  const int* __restrict__ dst,
                                                  const float* __restrict__ el,
                                                  const float* __restrict__ er,
                                                  float* __restrict__ ev,
                                                  int* __restrict__ mI,
                                                  int E) {
  int e = blockIdx.x * 256 + threadIdx.x;
  if (e >= E) return;
  float x = el[src[e]] + er[dst[e]];
  x = (x > 0.0f) ? x : NEG_SLOPE * x;
  ev[e] = x;
  atomicMax(&mI[dst[e]], f2ordered(x));
}

// ---- edge pass 2: exp(e - max) and per-dst sum -----------------------------
__global__ __launch_bounds__(256) void k_edge_exp(const int* __restrict__ dst,
                                                  float* __restrict__ ev,
                                                  const int* __restrict__ mI,
                                                  float* __restrict__ s,
                                                  int E) {
  int e = blockIdx.x * 256 + threadIdx.x;
  if (e >= E) return;
  int d = dst[e];
  float x = __expf(ev[e] - ordered2f(mI[d]));
  ev[e] = x;
  atomicAdd(&s[d], x);
}

// ---- edge pass 3: out[dst] += (ex/s[dst]) * feat[src]  (wave per edge) -----
__global__ __launch_bounds__(256) void k_edge_scatter(const int* __restrict__ src,
                                                      const int* __restrict__ dst,
                                                      const float* __restrict__ ev,
                                                      const float* __restrict__ s,
                                                      const float* __restrict__ F,
                                                      float* __restrict__ out,
                                                      int E) {
  int e    = blockIdx.x * 8 + (threadIdx.x >> 5);
  int lane = threadIdx.x & 31;
  if (e >= E) return;
  int sI = src[e], dI = dst[e];
  float alpha = ev[e] / s[dI];
  const float* f = F + (size_t)sI * DIM;
  float* o = out + (size_t)dI * DIM;
  atomicAdd(&o[lane],      alpha * f[lane]);
  atomicAdd(&o[lane + 32], alpha * f[lane + 32]);
}

// ---- column sums / sumsq for BatchNorm -------------------------------------
__global__ __launch_bounds__(256) void k_colstats(const float* __restrict__ X,
                                                  float* __restrict__ stats,
                                                  int n) {
  int col = threadIdx.x & 63;
  int r0  = blockIdx.x * 4 + (threadIdx.x >> 6);
  float sum = 0.0f, sq = 0.0f;
  for (int r = r0; r < n; r += gridDim.x * 4) {
    float v = X[(size_t)r * DIM + col];
    sum += v; sq += v * v;
  }
  atomicAdd(&stats[col], sum);
  atomicAdd(&stats[DIM + col], sq);
}

// ---- fused BatchNorm (+ optional ELU) --------------------------------------
__global__ __launch_bounds__(256) void k_bn(const float* __restrict__ X,
                                            const float* __restrict__ stats,
                                            const float* __restrict__ g,
                                            const float* __restrict__ beta,
                                            float* __restrict__ Y,
                                            int n, int do_elu) {
  int i = blockIdx.x * 256 + threadIdx.x;
  if (i >= n * DIM) return;
  int col = i & 63;
  float invN = 1.0f / (float)n;
  float mu  = stats[col] * invN;
  float var = fmaxf(stats[DIM + col] * invN - mu * mu, 0.0f);
  float rs  = rsqrtf(var + BN_EPS);
  float v = (X[i] - mu) * rs * g[col] + beta[col];
  if (do_elu) v = (v > 0.0f) ? v : (__expf(v) - 1.0f);
  Y[i] = v;
}

// ---------------------------------------------------------------------------
extern "C" void kernel_launch(void* const* d_in, const int* in_sizes, int n_in,
                              void* d_out, int out_size, void* d_ws, size_t ws_size,
                              hipStream_t stream) {
  (void)in_sizes; (void)n_in; (void)out_size; (void)ws_size;
  const float* node_w = (const float*)d_in[0];
  const int*   src    = (const int*)d_in[2];
  const int*   dst    = (const int*)d_in[3];

  char* p = (char*)d_ws;
  float* bufA = (float*)p; p += (size_t)N_NODES * DIM * 4;
  float* bufB = (float*)p; p += (size_t)N_NODES * DIM * 4;
  float* el   = (float*)p; p += (size_t)N_NODES * 4;
  float* er   = (float*)p; p += (size_t)N_NODES * 4;
  int*   mI   = (int*)p;   p += (size_t)N_NODES * 4;
  float* sD   = (float*)p; p += (size_t)N_NODES * 4;
  float* ev   = (float*)p; p += (size_t)N_EDGES * 4;
  float* st   = (float*)p; p += 2 * DIM * 4;

  auto layer = [&](const float* h, int li, float* feat, float* out, float* dest, int elu) {
    const float* W    = (const float*)d_in[4 + li * 6 + 0];
    const float* al   = (const float*)d_in[4 + li * 6 + 1];
    const float* ar   = (const float*)d_in[4 + li * 6 + 2];
    const float* g    = (const float*)d_in[4 + li * 6 + 4];
    const float* beta = (const float*)d_in[4 + li * 6 + 5];
    k_gemm64<<<(N_NODES + 127) / 128, 256, 0, stream>>>(h, W, feat, N_NODES);
    k_init<<<(N_NODES * DIM + 255) / 256, 256, 0, stream>>>(out, mI, sD, st, N_NODES);
    k_elr<<<(N_NODES + 7) / 8, 256, 0, stream>>>(feat, al, ar, el, er, N_NODES);
    k_edge_max<<<(N_EDGES + 255) / 256, 256, 0, stream>>>(src, dst, el, er, ev, mI, N_EDGES);
    k_edge_exp<<<(N_EDGES + 255) / 256, 256, 0, stream>>>(dst, ev, mI, sD, N_EDGES);
    k_edge_scatter<<<(N_EDGES + 7) / 8, 256, 0, stream>>>(src, dst, ev, sD, feat, out, N_EDGES);
    k_colstats<<<512, 256, 0, stream>>>(out, st, N_NODES);
    k_bn<<<(N_NODES * DIM + 255) / 256, 256, 0, stream>>>(out, st, g, beta, dest, N_NODES, elu);
  };

  // Buffer ping-pong: h/feat/out reuse is safe because kernels on `stream`
  // serialize, and each buffer is dead before it is overwritten.
  layer(node_w, 0, /*feat=*/bufB, /*out=*/bufA, /*dest=*/bufB, /*elu=*/1);
  layer(bufB,   1, /*feat=*/bufA, /*out=*/bufB, /*dest=*/bufA, /*elu=*/1);
  layer(bufA,   2, /*feat=*/bufB, /*out=*/bufA, /*dest=*/(float*)d_out, /*elu=*/0);
}